// EmbedBlock_47519518163429
// MI455X (gfx1250) — compile-verified
//
#include <hip/hip_runtime.h>
#include <hip/hip_bf16.h>
#include <stddef.h>

typedef __attribute__((ext_vector_type(2))) float v2f;
typedef __attribute__((ext_vector_type(8))) float v8f;

#define N_NODES 100000

__device__ __forceinline__ void atomic_add_f32(float* p, float v) {
    __hip_atomic_fetch_add(p, v, __ATOMIC_RELAXED, __HIP_MEMORY_SCOPE_AGENT);
}

// deg[i] = 1 (self loop), agg[i][0..3] = 0
__global__ __launch_bounds__(256) void gcn_init(unsigned int* __restrict__ deg,
                                                float4* __restrict__ agg, int n) {
    int i = blockIdx.x * blockDim.x + threadIdx.x;
    if (i >= n) return;
    deg[i] = 1u;
    agg[i] = make_float4(0.f, 0.f, 0.f, 0.f);
}

// deg[col[e]] += 1 over the 640k edges (native u32 atomic, L2-resident)
__global__ __launch_bounds__(256) void gcn_degree(const int* __restrict__ col,
                                                  unsigned int* __restrict__ deg, int E) {
    int e = blockIdx.x * blockDim.x + threadIdx.x;
    if (e >= E) return;
    __hip_atomic_fetch_add(&deg[col[e]], 1u, __ATOMIC_RELAXED, __HIP_MEMORY_SCOPE_AGENT);
}

// dis[i] = rsqrt(deg[i]) ; deg >= 1 always (self loop) so no guard needed
__global__ __launch_bounds__(256) void gcn_rsqrt(const unsigned int* __restrict__ deg,
                                                 float* __restrict__ dis, int n) {
    int i = blockIdx.x * blockDim.x + threadIdx.x;
    if (i >= n) return;
    dis[i] = rsqrtf((float)deg[i]);
}

// agg[col] += x[row] * (dis[row]*dis[col])   -- aggregation in 4-dim input space
__global__ __launch_bounds__(256) void gcn_edge_scatter(const int* __restrict__ row_idx,
                                                        const int* __restrict__ col_idx,
                                                        const float4* __restrict__ x4,
                                                        const float* __restrict__ dis,
                                                        float* __restrict__ agg, int E) {
    int e = blockIdx.x * blockDim.x + threadIdx.x;
    if (e >= E) return;
    int r = row_idx[e];
    int c = col_idx[e];
    float nrm = dis[r] * dis[c];
    float4 xv = x4[r];                 // global_load_b128, L2-resident (1.6 MB)
    float* ap = agg + (size_t)c * 4;
    atomic_add_f32(ap + 0, xv.x * nrm);
    atomic_add_f32(ap + 1, xv.y * nrm);
    atomic_add_f32(ap + 2, xv.z * nrm);
    atomic_add_f32(ap + 3, xv.w * nrm);
}

// out[16-row tile, :] = (agg + x*dis^2) @ W + b  via V_WMMA_F32_16X16X4_F32.
// One wave per 16-node tile; 8 column tiles of 16 channels each.
// A layout (ISA 32-bit A 16x4): lane L: row = L%16, VGPR0 = K=(L<16?0:2), VGPR1 = K+1.
// B layout (4x16, mirror):      lane L: col = L%16, VGPR0 = K=(L<16?0:2), VGPR1 = K+1.
// C/D layout: lane L: col = L%16 ; VGPR r holds row r + (L<16?0:8)  -> bias broadcast via C init.
__global__ __launch_bounds__(256) void gcn_gemm_wmma(const float* __restrict__ agg,
                                                     const float* __restrict__ x,
                                                     const float* __restrict__ dis,
                                                     const float* __restrict__ W,   // [4,128]
                                                     const float* __restrict__ bias,// [128]
                                                     float* __restrict__ out,       // [N,128]
                                                     int n_tiles) {
    const int lane = threadIdx.x & 31;
    const int wave = threadIdx.x >> 5;
    const int tile = blockIdx.x * 8 + wave;      // 8 waves / block
    if (tile >= n_tiles) return;                 // wave-uniform: EXEC stays all-1s for WMMA

    const int rloc  = lane & 15;
    const int khalf = (lane >> 4) << 1;          // 0 or 2
    const int row   = tile * 16 + rloc;

    const float d  = dis[row];
    const float sn = d * d;                      // self-loop norm = 1/deg
    v2f a;
    a.x = agg[(size_t)row * 4 + khalf]     + x[(size_t)row * 4 + khalf]     * sn;
    a.y = agg[(size_t)row * 4 + khalf + 1] + x[(size_t)row * 4 + khalf + 1] * sn;

    const int col   = lane & 15;
    const int rbase = tile * 16 + ((lane >> 4) << 3);   // rows 0..7 or 8..15 of tile

#pragma unroll
    for (int ct = 0; ct < 8; ++ct) {
        const int cbase = ct * 16;
        v2f bf;
        bf.x = W[(size_t)(khalf)     * 128 + cbase + col];
        bf.y = W[(size_t)(khalf + 1) * 128 + cbase + col];

        const float bv = bias[cbase + col];
        v8f c;
#pragma unroll
        for (int r = 0; r < 8; ++r) c[r] = bv;   // accumulate bias for free

        c = __builtin_amdgcn_wmma_f32_16x16x4_f32(
                /*neg_a=*/false, a, /*neg_b=*/false, bf,
                /*c_mod=*/(short)0, c, /*reuse_a=*/false, /*reuse_b=*/false);

        float* op = out + (size_t)rbase * 128 + cbase + col;
#pragma unroll
        for (int r = 0; r < 8; ++r) op[(size_t)r * 128] = c[r];
    }
}

extern "C" void kernel_launch(void* const* d_in, const int* in_sizes, int n_in,
                              void* d_out, int out_size, void* d_ws, size_t ws_size,
                              hipStream_t stream) {
    const float* x  = (const float*)d_in[0];        // [N,4]
    const int*   ei = (const int*)d_in[1];          // [2,E] (row; col)
    const float* W  = (const float*)d_in[2];        // [4,128]
    const float* b  = (const float*)d_in[3];        // [128]
    float* out = (float*)d_out;                     // [N,128]

    const int n = N_NODES;
    const int E = in_sizes[1] / 2;
    const int* row_idx = ei;
    const int* col_idx = ei + E;

    // workspace layout
    unsigned int* deg = (unsigned int*)d_ws;                    // N u32
    float*        dis = (float*)((char*)d_ws + (size_t)n * 4);  // N f32
    float*        agg = (float*)((char*)d_ws + (size_t)n * 8);  // N*4 f32

    const int TB = 256;
    gcn_init<<<(n + TB - 1) / TB, TB, 0, stream>>>(deg, (float4*)agg, n);
    gcn_degree<<<(E + TB - 1) / TB, TB, 0, stream>>>(col_idx, deg, E);
    gcn_rsqrt<<<(n + TB - 1) / TB, TB, 0, stream>>>(deg, dis, n);
    gcn_edge_scatter<<<(E + TB - 1) / TB, TB, 0, stream>>>(row_idx, col_idx,
                                                           (const float4*)x, dis, agg, E);

    const int n_tiles = n / 16;                     // 100000 / 16 = 6250 exactly
    const int blocks  = (n_tiles + 7) / 8;          // 8 waves per 256-thread block
    gcn_gemm_wmma<<<blocks, TB, 0, stream>>>(agg, x, dis, W, b, out, n_tiles);
}